// Rule_40767829574353
// MI455X (gfx1250) — compile-verified
//
#include <hip/hip_runtime.h>
#include <hip/hip_bf16.h>

// ---------------------------------------------------------------------------
// Ising-style stencil update, (B=8, C=3, H=2048, W=2048) f32, periodic wrap.
// Memory-bound: ~0.96 GB moved -> ~41 us at 23.3 TB/s. Strategy:
//  * one block per (batch,row); stage rows y-1,y,y+1 of the spin plane into
//    LDS via gfx1250 async global->LDS b128 copies (ASYNCcnt / s_wait_asynccnt)
//  * all other traffic as coalesced b128 loads/stores
// ---------------------------------------------------------------------------

#define AS_GLOBAL __attribute__((address_space(1)))
#define AS_SHARED __attribute__((address_space(3)))

typedef int vint4 __attribute__((ext_vector_type(4)));

#if defined(__has_builtin)
#if __has_builtin(__builtin_amdgcn_global_load_async_to_lds_b128)
#define HAVE_ASYNC_LDS 1
#endif
#endif
#ifndef HAVE_ASYNC_LDS
#define HAVE_ASYNC_LDS 0
#endif

static constexpr int kW = 2048;
static constexpr int kH = 2048;
static constexpr int kC = 3;
static constexpr int kThreads = 256;

// Copy one full row (2048 f32 = 512 x b128) from global to LDS.
// 256 threads -> 2 async b128 per thread.
__device__ __forceinline__ void async_copy_row(const float* __restrict__ g,
                                               float* l) {
  const int t = (int)threadIdx.x;
#pragma unroll
  for (int i = 0; i < 2; ++i) {
    const int off = (t + i * kThreads) * 4;  // float index, 16B aligned
#if HAVE_ASYNC_LDS
    __builtin_amdgcn_global_load_async_to_lds_b128(
        (AS_GLOBAL vint4*)(g + off), (AS_SHARED vint4*)(l + off),
        /*imm_offset=*/0, /*cpol=*/0);
#else
    *(float4*)(l + off) = *(const float4*)(g + off);
#endif
  }
}

__device__ __forceinline__ void wait_async_then_barrier() {
#if HAVE_ASYNC_LDS
#if defined(__has_builtin) && __has_builtin(__builtin_amdgcn_s_wait_asynccnt)
  __builtin_amdgcn_s_wait_asynccnt(0);
#else
  asm volatile("s_wait_asynccnt 0" ::: "memory");
#endif
#endif
  __syncthreads();
}

__global__ __launch_bounds__(kThreads) void ising_stencil_kernel(
    const float* __restrict__ x,     // (B,3,H,W)
    const float* __restrict__ rnd,   // (B,1,H,W)
    const float* __restrict__ drop,  // (H,W)
    const float* __restrict__ w9,    // (9,)
    float* __restrict__ out) {       // (B,3,H,W)
  const int y = (int)blockIdx.x;
  const int b = (int)blockIdx.y;
  const size_t HW = (size_t)kH * kW;

  const float* sP  = x + ((size_t)b * kC + 0) * HW;
  const float* trP = x + ((size_t)b * kC + 1) * HW;
  const float* bP  = x + ((size_t)b * kC + 2) * HW;
  const float* rP  = rnd + (size_t)b * HW;

  float* oS = out + ((size_t)b * kC + 0) * HW;
  float* oT = out + ((size_t)b * kC + 1) * HW;
  float* oB = out + ((size_t)b * kC + 2) * HW;

  __shared__ float sm[3 * kW];
  float* rU = sm;            // row y-1 (wrapped)
  float* rM = sm + kW;       // row y
  float* rD = sm + 2 * kW;   // row y+1 (wrapped)

  const int ym1 = (y + kH - 1) & (kH - 1);
  const int yp1 = (y + 1) & (kH - 1);

  async_copy_row(sP + (size_t)ym1 * kW, rU);
  async_copy_row(sP + (size_t)y   * kW, rM);
  async_copy_row(sP + (size_t)yp1 * kW, rD);

  // Uniform 3x3 stencil weights (scalar loads; hide under async copies).
  const float w00 = w9[0], w01 = w9[1], w02 = w9[2];
  const float w10 = w9[3], w11 = w9[4], w12 = w9[5];
  const float w20 = w9[6], w21 = w9[7], w22 = w9[8];

  wait_async_then_barrier();

  const int t = (int)threadIdx.x;
#pragma unroll
  for (int g = 0; g < 2; ++g) {
    const int x0 = (t + g * kThreads) * 4;
    const int xl = (x0 - 1) & (kW - 1);
    const int xr = (x0 + 4) & (kW - 1);

    const float4 u4 = *(const float4*)(rU + x0);
    const float4 m4 = *(const float4*)(rM + x0);
    const float4 d4 = *(const float4*)(rD + x0);

    // columns x0-1 .. x0+4 of each stencil row
    const float u[6] = {rU[xl], u4.x, u4.y, u4.z, u4.w, rU[xr]};
    const float m[6] = {rM[xl], m4.x, m4.y, m4.z, m4.w, rM[xr]};
    const float d[6] = {rD[xl], d4.x, d4.y, d4.z, d4.w, rD[xr]};

    const size_t off = (size_t)y * kW + x0;
    const float4 tr4 = *(const float4*)(trP + off);
    const float4 bb4 = *(const float4*)(bP + off);
    const float4 rn4 = *(const float4*)(rP + off);
    const float4 dr4 = *(const float4*)(drop + off);

    const float trv[4] = {tr4.x, tr4.y, tr4.z, tr4.w};
    const float bbv[4] = {bb4.x, bb4.y, bb4.z, bb4.w};
    const float rnv[4] = {rn4.x, rn4.y, rn4.z, rn4.w};
    const float drv[4] = {dr4.x, dr4.y, dr4.z, dr4.w};

    float osv[4], otv[4];
#pragma unroll
    for (int j = 0; j < 4; ++j) {
      const float s = m[j + 1];
      // Js[y][x] = sum_ij w[i][j] * s[y+i-1][x+j-1]  (periodic)
      float Js = w00 * u[j] + w01 * u[j + 1] + w02 * u[j + 2];
      Js += w10 * m[j] + w11 * s + w12 * m[j + 2];
      Js += w20 * d[j] + w21 * d[j + 1] + w22 * d[j + 2];
      const float de = 2.0f * s * Js;
      const float p = (de <= 0.0f) ? 1.0f : __expf(-de * bbv[j]);
      const bool fl = (rnv[j] < p) && (drv[j] > 0.9f);
      osv[j] = fl ? -s : s;
      otv[j] = fmaf(0.995f, trv[j], 0.005f * s);
    }

    *(float4*)(oS + off) = make_float4(osv[0], osv[1], osv[2], osv[3]);
    *(float4*)(oT + off) = make_float4(otv[0], otv[1], otv[2], otv[3]);
    *(float4*)(oB + off) = bb4;  // passthrough channel
  }
}

extern "C" void kernel_launch(void* const* d_in, const int* in_sizes, int n_in,
                              void* d_out, int out_size, void* d_ws,
                              size_t ws_size, hipStream_t stream) {
  (void)n_in;
  (void)out_size;
  (void)d_ws;
  (void)ws_size;
  const float* x    = (const float*)d_in[0];  // (B,3,H,W)
  const float* rnd  = (const float*)d_in[1];  // (B,1,H,W)
  const float* drop = (const float*)d_in[2];  // (H,W)
  const float* w9   = (const float*)d_in[3];  // (9,)
  float* out = (float*)d_out;

  const int B = in_sizes[1] / (kH * kW);  // rand is (B,1,H,W)
  dim3 grid(kH, (unsigned)B);
  dim3 block(kThreads);
  ising_stencil_kernel<<<grid, block, 0, stream>>>(x, rnd, drop, w9, out);
}